// DynaMixerOp_w_50586124812904
// MI455X (gfx1250) — compile-verified
//
#include <hip/hip_runtime.h>

typedef __attribute__((ext_vector_type(2))) float v2f;
typedef __attribute__((ext_vector_type(8))) float v8f;

// V_WMMA_F32_16X16X4_F32: D(16x16 f32) = A(16x4) x B(4x16) + C
static __device__ inline v8f wmma4(v2f a, v2f b, v8f c) {
  return __builtin_amdgcn_wmma_f32_16x16x4_f32(
      false, a, false, b, (short)0, c, false, false);
}

#define ZV8 {0.f,0.f,0.f,0.f,0.f,0.f,0.f,0.f}

#if __has_builtin(__builtin_amdgcn_global_load_async_to_lds_b128)
#define USE_ASYNC_LDS 1
typedef int v4i __attribute__((vector_size(16)));
typedef __attribute__((address_space(1))) v4i g_v4i;  // global (AS1)
typedef __attribute__((address_space(3))) v4i l_v4i;  // LDS    (AS3)
#else
#define USE_ASYNC_LDS 0
#endif

static __device__ inline void wait_async0() {
#if __has_builtin(__builtin_amdgcn_s_wait_asynccnt)
  __builtin_amdgcn_s_wait_asynccnt(0);
#else
  asm volatile("s_wait_asynccnt 0" ::: "memory");
#endif
}

// ---------------------------------------------------------------------------
// Kernel A: one block per (batch, h-half).  h is a pure batch dim in stages
// 1-3, so each block handles 16 h values: Y[16][512] -> Z[128][16] -> softmax.
// attn layout: attn[b][h][u][v], u = 4g + w1l.
// ---------------------------------------------------------------------------
__global__ __launch_bounds__(256) void dynamixer_logits_softmax(
    const float* __restrict__ x,    // [128][32][32][384]  (b,h,w,d)
    const float* __restrict__ W2d,  // [16][384]
    const float* __restrict__ b2d,  // [16]
    const float* __restrict__ W1d,  // [1024][64]
    const float* __restrict__ b1d,  // [1024]
    float* __restrict__ attn)       // [128][32][32][32]
{
  extern __shared__ float lds[];
  float* Y = lds;            // [16][512], Y[e][w*16 + hl]
  float* Z = lds + 16 * 512; // [128][16], Z[o][hl]

  const int bb   = blockIdx.x;
  const int b    = bb >> 1;
  const int h0b  = (bb & 1) * 16;  // h-half base
  const int tid  = threadIdx.x;
  const int lane = tid & 31;
  const int wid  = tid >> 5;
  const int lh   = lane & 15;
  const int hi   = lane >> 4;

  const float* xb = x + (size_t)b * (32 * 32 * 384);

  // ---- Stage 1: Y[e][w*16+hl] = sum_d W2d[e,d]*x[b, h0b+hl, w, d] + b2d[e]
  // GEMM M=16(e), N=512(w,hl), K=384(d). 32 N-tiles, 4 per wave; tile tt -> w.
  for (int t = 0; t < 4; ++t) {
    const int tt = wid * 4 + t;        // == w for this tile
    const int n  = tt * 16 + lh;
    const float* col = xb + ((size_t)(h0b + lh) * 32 + tt) * 384;  // x[b,h,w,:]
    __builtin_prefetch(col + 384, 0, 3);  // next tile's column (w+1)
    v8f c = ZV8;
#pragma unroll 4
    for (int k = 0; k < 384; k += 4) {
      const int kk = k + hi * 2;
      v2f a   = *(const v2f*)(W2d + lh * 384 + kk);  // A[M=lh, kk..kk+1]
      v2f bb2 = *(const v2f*)(col + kk);             // B[kk..kk+1, N=n]
      c = wmma4(a, bb2, c);
    }
    const int mbase = hi * 8;
#pragma unroll
    for (int r = 0; r < 8; ++r)
      Y[(mbase + r) * 512 + n] = c[r] + b2d[mbase + r];
  }
  __syncthreads();

  // ---- Stage 2+3 per group: Z = Wg(128x64) @ Yg(64x16); softmax over w2.
  // Yg[c][hl] == Y[2g*512 + c*16 + hl]  (contiguous [64][16] block).
  for (int g = 0; g < 8; ++g) {
    const float* Yg = Y + g * 1024;
    const int o0 = wid * 16;  // each wave owns one 16-row o-tile
    const float* Wrow = W1d + (size_t)(g * 128 + o0 + lh) * 64;
    v8f c0 = ZV8;
#pragma unroll 4
    for (int k = 0; k < 64; k += 4) {
      const int kk = k + hi * 2;
      v2f a = *(const v2f*)(Wrow + kk);  // A[M=o0+lh, kk..kk+1]
      v2f b0;
      b0.x = Yg[kk * 16 + lh];
      b0.y = Yg[(kk + 1) * 16 + lh];
      c0 = wmma4(a, b0, c0);
    }
#pragma unroll
    for (int r = 0; r < 8; ++r)
      Z[(o0 + hi * 8 + r) * 16 + lh] = c0[r];
    __syncthreads();

    // softmax rows: (w1l in [0,4), hl in [0,16)) over w2, o = w1l*32 + w2
    if (tid < 64) {
      const int w1l = tid >> 4;
      const int hl  = tid & 15;
      const int u   = g * 4 + w1l;
      const float* brow = b1d + u * 32;
      float v[32];
      float m = -3.4e38f;
#pragma unroll
      for (int w2 = 0; w2 < 32; ++w2) {
        v[w2] = Z[(w1l * 32 + w2) * 16 + hl] + brow[w2];
        m = fmaxf(m, v[w2]);
      }
      float s = 0.f;
#pragma unroll
      for (int w2 = 0; w2 < 32; ++w2) {
        v[w2] = __expf(v[w2] - m);
        s += v[w2];
      }
      const float inv = 1.f / s;
      float* arow = attn + (((size_t)b * 32 + (h0b + hl)) * 32 + u) * 32;
#pragma unroll
      for (int w2 = 0; w2 < 32; w2 += 4) {
        float4 o4 = make_float4(v[w2] * inv, v[w2 + 1] * inv,
                                v[w2 + 2] * inv, v[w2 + 3] * inv);
        *(float4*)(arow + w2) = o4;
      }
    }
    __syncthreads();  // protect Z before next group overwrites it
  }
}

// ---------------------------------------------------------------------------
// Kernel B: per (b,h). out[b,h] (32x384) = attn[b,h](32x32) @ x[b,h](32x384).
// x staged into LDS once (single HBM pass over x here), via async-to-LDS
// loads (ASYNCcnt) when the toolchain exposes them.
// ---------------------------------------------------------------------------
__global__ __launch_bounds__(256) void dynamixer_mix(
    const float* __restrict__ x,     // [4096][32][384] per (b,h)
    const float* __restrict__ attn,  // [4096][32][32]
    float* __restrict__ out)         // [4096][32][384]
{
  __shared__ float xs[32 * 384];  // 48 KB
  __shared__ float as[32 * 32];   // 4 KB

  const int bh   = blockIdx.x;
  const int tid  = threadIdx.x;
  const int lane = tid & 31;
  const int wid  = tid >> 5;
  const int lh   = lane & 15;
  const int hi   = lane >> 4;

  const float* xbh = x    + (size_t)bh * (32 * 384);
  const float* abh = attn + (size_t)bh * 1024;

#if USE_ASYNC_LDS
#pragma unroll
  for (int i = tid; i < (32 * 384) / 4; i += 256)
    __builtin_amdgcn_global_load_async_to_lds_b128(
        (g_v4i*)(xbh + i * 4), (l_v4i*)(xs + i * 4), 0, 0);
  __builtin_amdgcn_global_load_async_to_lds_b128(
      (g_v4i*)(abh + tid * 4), (l_v4i*)(as + tid * 4), 0, 0);
  wait_async0();
#else
#pragma unroll
  for (int i = tid; i < (32 * 384) / 4; i += 256)
    *(float4*)(xs + i * 4) = *(const float4*)(xbh + i * 4);
  *(float4*)(as + tid * 4) = *(const float4*)(abh + tid * 4);
#endif
  __syncthreads();

  float* obh = out + (size_t)bh * (32 * 384);

  // 24 N-tiles of 16 (d), 2 M-tiles (u). Each wave: 3 N-tiles x both M-tiles.
  for (int j = 0; j < 3; ++j) {
    const int n0 = (wid * 3 + j) * 16;
    const int d  = n0 + lh;
    v8f c0 = ZV8, c1 = ZV8;
#pragma unroll
    for (int k = 0; k < 32; k += 4) {
      const int kk = k + hi * 2;
      v2f bf;
      bf.x = xs[kk * 384 + d];
      bf.y = xs[(kk + 1) * 384 + d];
      v2f a0 = *(const v2f*)(as + lh * 32 + kk);         // rows u = 0..15
      v2f a1 = *(const v2f*)(as + (16 + lh) * 32 + kk);  // rows u = 16..31
      c0 = wmma4(a0, bf, c0);
      c1 = wmma4(a1, bf, c1);
    }
#pragma unroll
    for (int r = 0; r < 8; ++r) {
      obh[(hi * 8 + r) * 384 + d]      = c0[r];
      obh[(16 + hi * 8 + r) * 384 + d] = c1[r];
    }
  }
}

extern "C" void kernel_launch(void* const* d_in, const int* in_sizes, int n_in,
                              void* d_out, int out_size, void* d_ws, size_t ws_size,
                              hipStream_t stream) {
  const float* x   = (const float*)d_in[0];
  const float* W2d = (const float*)d_in[1];
  const float* b2d = (const float*)d_in[2];
  const float* W1d = (const float*)d_in[3];
  const float* b1d = (const float*)d_in[4];
  float* attn = (float*)d_ws;      // 128*32*32*32 floats = 16.8 MB
  float* out  = (float*)d_out;

  const size_t shmemA = (16 * 512 + 128 * 16) * sizeof(float);  // 40 KB
  hipLaunchKernelGGL(dynamixer_logits_softmax, dim3(256), dim3(256), shmemA,
                     stream, x, W2d, b2d, W1d, b1d, attn);
  hipLaunchKernelGGL(dynamixer_mix, dim3(128 * 32), dim3(256), 0, stream,
                     x, attn, out);
}